// GCMC_56899726737489
// MI455X (gfx1250) — compile-verified
//
#include <hip/hip_runtime.h>

// ---- problem constants (match reference) ----
#define N_USERS  100000
#define N_ITEMS  50000
#define N_NODESC 150000
#define N_RELC   5
#define NNZ_E    2000000
#define DIM      64
#define B_EVALC  16384
#define SLOTS    (2 * B_EVALC)   // 32768 compact rows

typedef float v2f __attribute__((ext_vector_type(2)));
typedef float v8f __attribute__((ext_vector_type(8)));

// ---------------------------------------------------------------------------
// map[node] = compact slot (or -1). Duplicated users/items: last-writer wins;
// all readers go through map so they all see the winning slot (consistent).
// ---------------------------------------------------------------------------
__global__ void k_init_map(int* __restrict__ map) {
    int i = blockIdx.x * blockDim.x + threadIdx.x;
    if (i < N_NODESC) map[i] = -1;
}

__global__ void k_fill_map(const int* __restrict__ users,
                           const int* __restrict__ items,
                           int* __restrict__ map) {
    int b = blockIdx.x * blockDim.x + threadIdx.x;
    if (b < B_EVALC) {
        map[users[b]]           = b;
        map[N_USERS + items[b]] = B_EVALC + b;
    }
}

__global__ void k_zero4(float4* __restrict__ p, int n4) {
    int i = blockIdx.x * blockDim.x + threadIdx.x;
    if (i < n4) p[i] = make_float4(0.f, 0.f, 0.f, 0.f);
}

// ---------------------------------------------------------------------------
// COO scatter: one wave32 per edge. Lanes cooperatively gather the 64-float
// (256B) embedding row coalesced (float2/lane) and atomically add into the
// compact agg row. Edges whose destination is not a selected node are skipped
// before touching cols/vals/E (saves ~78% of gather+atomic traffic).
// ---------------------------------------------------------------------------
__global__ void k_scatter(const int*   __restrict__ rows,
                          const int*   __restrict__ cols,
                          const float* __restrict__ vals,
                          const int*   __restrict__ map,
                          const float* __restrict__ uemb,
                          const float* __restrict__ iemb,
                          float*       __restrict__ agg) {
    const int lane = threadIdx.x & 31;                       // wave32
    const int wave = (blockIdx.x * blockDim.x + threadIdx.x) >> 5;
    const int nw   = (gridDim.x * blockDim.x) >> 5;
    for (int e = wave; e < NNZ_E; e += nw) {
        __builtin_prefetch(rows + e + nw, 0, 0);             // global_prefetch_b8
        int s = map[rows[e]];
        if (s < 0) continue;
        int   c = cols[e];
        float v = vals[e];
        const float* src = (c < N_USERS) ? (uemb + (size_t)c * DIM)
                                         : (iemb + (size_t)(c - N_USERS) * DIM);
        v2f ev = ((const v2f*)src)[lane];
        atomicAdd(agg + (size_t)s * DIM + 2 * lane,     ev.x * v);
        atomicAdd(agg + (size_t)s * DIM + 2 * lane + 1, ev.y * v);
    }
}

// ---------------------------------------------------------------------------
// Fused per-relation dense stage (one wave per 16-row strip, fp32 WMMA):
//   H  = leaky_relu(agg[16x64] @ W[64x64] + bias, 0.2)      (WMMA chain #1)
//   logits[16x64] += H @ AW[64x64]                          (WMMA chain #2)
// H is re-laid-out C-layout -> A-layout through 4KB/wave of LDS.
// ---------------------------------------------------------------------------
__global__ void __launch_bounds__(256)
k_fused_gemm(const float* __restrict__ agg,
             const float* __restrict__ W,      // [64][64] row-major (d x j)
             const float* __restrict__ bias,   // [64]
             const float* __restrict__ AW,     // [64][64] row-major (j x d)
             float*       __restrict__ logits) {
    __shared__ float smem[8 * 16 * 64];        // 32 KB: 16x64 H tile per wave

    const int lane = threadIdx.x & 31;
    const int wave = threadIdx.x >> 5;
    const int row0 = (blockIdx.x * 8 + wave) * 16;
    const int lo   = lane & 15;
    const int hi   = lane >> 4;                // 0 / 1
    const int k2   = hi * 2;                   // K sub-offset within K=4 step

    // ---- GEMM 1: H = agg @ W ----
    v8f acc[4];
    #pragma unroll
    for (int t = 0; t < 4; ++t) acc[t] = {};
    const float* arow = agg + (size_t)(row0 + lo) * DIM;     // A: M = lo
    #pragma unroll
    for (int kk = 0; kk < 16; ++kk) {
        const int kb = kk * 4 + k2;
        v2f a; a.x = arow[kb]; a.y = arow[kb + 1];
        #pragma unroll
        for (int t = 0; t < 4; ++t) {
            const int n = t * 16 + lo;
            v2f b; b.x = W[(size_t)kb * DIM + n];
                   b.y = W[(size_t)(kb + 1) * DIM + n];
            acc[t] = __builtin_amdgcn_wmma_f32_16x16x4_f32(
                false, a, false, b, (short)0, acc[t], false, false);
        }
    }

    // ---- bias + leaky_relu(0.2), stage H into LDS (A-layout source) ----
    float* h = smem + wave * (16 * 64);
    #pragma unroll
    for (int t = 0; t < 4; ++t) {
        const int   j  = t * 16 + lo;
        const float bj = bias[j];
        #pragma unroll
        for (int v = 0; v < 8; ++v) {
            float x = acc[t][v] + bj;
            x = (x > 0.f) ? x : 0.2f * x;
            h[(v + 8 * hi) * 64 + j] = x;      // row M = v + 8*hi, col j
        }
    }
    __syncthreads();

    // ---- GEMM 2: logits += H @ AW (accumulator preloaded from logits) ----
    v8f acc2[4];
    #pragma unroll
    for (int t = 0; t < 4; ++t)
        #pragma unroll
        for (int v = 0; v < 8; ++v)
            acc2[t][v] = logits[(size_t)(row0 + v + 8 * hi) * DIM + t * 16 + lo];

    #pragma unroll
    for (int kk = 0; kk < 16; ++kk) {
        const int kb = kk * 4 + k2;
        v2f a; a.x = h[lo * 64 + kb]; a.y = h[lo * 64 + kb + 1];
        #pragma unroll
        for (int t = 0; t < 4; ++t) {
            const int n = t * 16 + lo;
            v2f b; b.x = AW[(size_t)kb * DIM + n];
                   b.y = AW[(size_t)(kb + 1) * DIM + n];
            acc2[t] = __builtin_amdgcn_wmma_f32_16x16x4_f32(
                false, a, false, b, (short)0, acc2[t], false, false);
        }
    }

    #pragma unroll
    for (int t = 0; t < 4; ++t)
        #pragma unroll
        for (int v = 0; v < 8; ++v)
            logits[(size_t)(row0 + v + 8 * hi) * DIM + t * 16 + lo] = acc2[t][v];
}

// ---------------------------------------------------------------------------
// out[b] = dot(logits[map[u]] + affine_b, logits[map[N_USERS+i]] + affine_b)
// ---------------------------------------------------------------------------
__global__ void k_out(const int*   __restrict__ map,
                      const float* __restrict__ logits,
                      const int*   __restrict__ users,
                      const int*   __restrict__ items,
                      const float* __restrict__ ab,
                      float*       __restrict__ out) {
    int b = blockIdx.x * blockDim.x + threadIdx.x;
    if (b >= B_EVALC) return;
    const float* Lu = logits + (size_t)map[users[b]] * DIM;
    const float* Li = logits + (size_t)map[N_USERS + items[b]] * DIM;
    float s = 0.f;
    #pragma unroll
    for (int d = 0; d < DIM; ++d)
        s += (Lu[d] + ab[d]) * (Li[d] + ab[d]);
    out[b] = s;
}

// ---------------------------------------------------------------------------
extern "C" void kernel_launch(void* const* d_in, const int* in_sizes, int n_in,
                              void* d_out, int out_size, void* d_ws, size_t ws_size,
                              hipStream_t stream) {
    const int*   users    = (const int*)  d_in[0];
    const int*   items    = (const int*)  d_in[1];
    const int*   rows     = (const int*)  d_in[2];
    const int*   cols     = (const int*)  d_in[3];
    const float* vals     = (const float*)d_in[4];
    const float* uemb     = (const float*)d_in[5];
    const float* iemb     = (const float*)d_in[6];
    const float* W_rel    = (const float*)d_in[7];
    const float* b_rel    = (const float*)d_in[8];
    const float* affine_W = (const float*)d_in[9];
    const float* affine_b = (const float*)d_in[10];
    float* out = (float*)d_out;

    // workspace layout: map (600KB) | agg (8.4MB) | logits (8.4MB)
    char*  ws  = (char*)d_ws;
    int*   map = (int*)ws;
    size_t off = (((size_t)N_NODESC * sizeof(int)) + 255) & ~(size_t)255;
    float* agg    = (float*)(ws + off);
    float* logits = agg + (size_t)SLOTS * DIM;

    k_init_map<<<(N_NODESC + 255) / 256, 256, 0, stream>>>(map);
    k_fill_map<<<(B_EVALC + 255) / 256, 256, 0, stream>>>(users, items, map);
    k_zero4<<<(SLOTS * DIM / 4 + 255) / 256, 256, 0, stream>>>((float4*)logits,
                                                               SLOTS * DIM / 4);

    for (int k = 0; k < N_RELC; ++k) {
        k_zero4<<<(SLOTS * DIM / 4 + 255) / 256, 256, 0, stream>>>((float4*)agg,
                                                                   SLOTS * DIM / 4);
        k_scatter<<<2048, 256, 0, stream>>>(rows + (size_t)k * NNZ_E,
                                            cols + (size_t)k * NNZ_E,
                                            vals + (size_t)k * NNZ_E,
                                            map, uemb, iemb, agg);
        k_fused_gemm<<<SLOTS / 16 / 8, 256, 0, stream>>>(
            agg,
            W_rel    + (size_t)k * DIM * DIM,
            b_rel    + (size_t)k * DIM,
            affine_W + (size_t)k * DIM * DIM,
            logits);
    }

    k_out<<<(B_EVALC + 255) / 256, 256, 0, stream>>>(map, logits, users, items,
                                                     affine_b, out);
}